// OperatorWrapperBlock_65292092834407
// MI455X (gfx1250) — compile-verified
//
#include <hip/hip_runtime.h>
#include <hip/hip_bf16.h>
#include <math.h>

// Problem constants (match reference)
#define GH   128
#define GW   128
#define CIN  4
#define COUT 4
#define CHID 16
#define BATCH 16
#define NPTS 262144   // 2^18

typedef __attribute__((ext_vector_type(2))) float v2f;
typedef __attribute__((ext_vector_type(8))) float v8f;

// ---------------------------------------------------------------------------
// 0) zero the scatter accumulators (seg + cnt are contiguous in ws)
// ---------------------------------------------------------------------------
__global__ void fno_zero_kernel(float* __restrict__ p, int n) {
    int i = blockIdx.x * blockDim.x + threadIdx.x;
    if (i < n) p[i] = 0.0f;
}

// ---------------------------------------------------------------------------
// 1) points_to_grid scatter: atomicAdd u into per-cell sums + counts
//    seg layout: (B, GH*GW, CIN) NHWC ; cnt: (B, GH*GW)
// ---------------------------------------------------------------------------
__global__ void fno_scatter_kernel(const float* __restrict__ u,
                                   const float* __restrict__ coords,
                                   float* __restrict__ seg,
                                   float* __restrict__ cnt) {
    int idx = blockIdx.x * blockDim.x + threadIdx.x;   // 0 .. B*N-1
    if (idx >= BATCH * NPTS) return;
    int b = idx >> 18;                                 // N = 2^18
    float c0 = coords[2 * idx + 0];
    float c1 = coords[2 * idx + 1];
    int i = (int)(fminf(fmaxf(c0, 0.0f), 1.0f) * (float)(GH - 1));
    int j = (int)(fminf(fmaxf(c1, 0.0f), 1.0f) * (float)(GW - 1));
    int cell = (b << 14) + i * GW + j;                 // GH*GW = 16384
    const float* up = u + (size_t)idx * CIN;
    atomicAdd(&seg[(size_t)cell * CIN + 0], up[0]);
    atomicAdd(&seg[(size_t)cell * CIN + 1], up[1]);
    atomicAdd(&seg[(size_t)cell * CIN + 2], up[2]);
    atomicAdd(&seg[(size_t)cell * CIN + 3], up[3]);
    atomicAdd(&cnt[cell], 1.0f);
}

// ---------------------------------------------------------------------------
// 2) normalize: grid_in = seg / max(cnt, 1)   (in place on seg)
// ---------------------------------------------------------------------------
__global__ void fno_normalize_kernel(float* __restrict__ seg,
                                     const float* __restrict__ cnt) {
    int cell = blockIdx.x * blockDim.x + threadIdx.x;  // 0 .. B*16384-1
    if (cell >= BATCH * GH * GW) return;
    float inv = 1.0f / fmaxf(cnt[cell], 1.0f);
    float* p = seg + (size_t)cell * CIN;
    p[0] *= inv; p[1] *= inv; p[2] *= inv; p[3] *= inv;
}

// ---------------------------------------------------------------------------
// 3) conv1 (4 -> 16 ch, 3x3 SAME) + bias + exact GELU via f32 WMMA 16x16x4.
//    Block = one (batch,row). Halo rows staged in LDS (zero padded), weights
//    staged transposed; tap loop is branch-free ds_load -> wmma.
//    grid_in: (B,GH,GW,4) NHWC ; hid: (B,GH,GW,16) NHWC
//    w1: (16,4,3,3) ; b1: (16)
// ---------------------------------------------------------------------------
__global__ void fno_conv1_wmma_kernel(const float* __restrict__ gin,
                                      const float* __restrict__ w1,
                                      const float* __restrict__ b1,
                                      float* __restrict__ hid) {
    __shared__ float sIn[3][132][CIN];   // cols 0..129 used = x -1..128
    __shared__ float sW[9][16][4];       // [tap][n][k]

    int blk = blockIdx.x;                // 0..2047
    int y = blk & 127;
    int b = blk >> 7;
    int tid = threadIdx.x;

    // stage weights transposed: w1 flat f = ((n*4+k)*9 + ti)
    for (int f = tid; f < 16 * 4 * 9; f += 256) {
        int n  = f / 36;
        int k  = (f / 9) & 3;
        int ti = f % 9;
        sW[ti][n][k] = w1[f];
    }
    // stage input halo: 3 rows x 130 cols, one float4 (pixel) each
    for (int e = tid; e < 3 * 130; e += 256) {
        int r  = e / 130;
        int c  = e % 130;                // storage col = x + 1
        int yy = y - 1 + r;
        int xx = c - 1;
        float4 v = make_float4(0.f, 0.f, 0.f, 0.f);
        if (yy >= 0 && yy < GH && xx >= 0 && xx < GW)
            v = ((const float4*)gin)[(size_t)(b * GH + yy) * GW + xx];
        *(float4*)&sIn[r][c][0] = v;
    }
    __syncthreads();

    int lane = tid & 31;
    int wv   = tid >> 5;                 // wave = tile index 0..7
    int x0   = wv * 16;
    int m    = lane & 15;                // A row (pixel)
    int kb   = (lane >> 4) * 2;          // K pair: 0 or 2
    int n    = lane & 15;                // B/D column (hidden channel)

    v8f acc = {};
    #pragma unroll
    for (int ti = 0; ti < 9; ++ti) {
        int dy = ti / 3;                 // 0..2 == rows y-1..y+1
        int dx = ti % 3;                 // 0..2, storage col has +1 built in
        const float* pa = &sIn[dy][x0 + m + dx][kb];
        v2f a;  a.x = pa[0]; a.y = pa[1];
        const float* pw = &sW[ti][n][kb];
        v2f w;  w.x = pw[0]; w.y = pw[1];
        acc = __builtin_amdgcn_wmma_f32_16x16x4_f32(
            false, a, false, w, (short)0, acc, false, false);
    }

    float bias = b1[n];
    int mtop = (lane >> 4) * 8;          // D: VGPR v -> M = v + 8*(lane>=16)
    #pragma unroll
    for (int v = 0; v < 8; ++v) {
        float x = acc[v] + bias;
        float g = 0.5f * x * (1.0f + erff(x * 0.70710678118654752f)); // exact GELU
        int mm = mtop + v;
        hid[((size_t)(b * GH + y) * GW + (x0 + mm)) * CHID + n] = g;
    }
}

// ---------------------------------------------------------------------------
// 4) conv2 (16 -> 4 ch, 3x3 SAME) + bias via f32 WMMA 16x16x4 (36 per tile).
//    Pixel stride in LDS padded to 20 floats (bank-conflict-free, 16B aligned).
//    Weights zero-padded to n=16 in LDS -> branch-free K loop.
//    hid: (B,GH,GW,16) NHWC ; gout: (B,GH,GW,4) NHWC ; w2: (4,16,3,3) ; b2: (4)
// ---------------------------------------------------------------------------
#define HSTR 20
__global__ void fno_conv2_wmma_kernel(const float* __restrict__ hid,
                                      const float* __restrict__ w2,
                                      const float* __restrict__ b2,
                                      float* __restrict__ gout) {
    __shared__ float sH[3][132][HSTR];   // cols 0..129 used; 16 of 20 floats used
    __shared__ float sW2[9][16][16];     // [tap][n(pad 16)][k] ; n>=4 zeros

    int blk = blockIdx.x;
    int y = blk & 127;
    int b = blk >> 7;
    int tid = threadIdx.x;

    // stage padded weights: element (ti, n, k); w2 flat f = ((n*16+k)*9 + ti)
    for (int e = tid; e < 9 * 16 * 16; e += 256) {
        int ti = e >> 8;
        int n  = (e >> 4) & 15;
        int k  = e & 15;
        float v = 0.0f;
        if (n < COUT) v = w2[((n * CHID + k) * 9) + ti];
        sW2[ti][n][k] = v;
    }
    // stage hidden halo: 3 rows x 130 cols x 4 float4s
    for (int e = tid; e < 3 * 130 * 4; e += 256) {
        int r   = e / 520;
        int rem = e % 520;
        int c   = rem >> 2;              // storage col = x + 1
        int q   = rem & 3;               // which float4 of the 16-ch pixel
        int yy  = y - 1 + r;
        int xx  = c - 1;
        float4 v = make_float4(0.f, 0.f, 0.f, 0.f);
        if (yy >= 0 && yy < GH && xx >= 0 && xx < GW)
            v = ((const float4*)hid)[((size_t)(b * GH + yy) * GW + xx) * 4 + q];
        *(float4*)&sH[r][c][4 * q] = v;
    }
    __syncthreads();

    int lane = tid & 31;
    int wv   = tid >> 5;
    int x0   = wv * 16;
    int m    = lane & 15;
    int kb   = (lane >> 4) * 2;
    int n    = lane & 15;

    v8f acc = {};
    #pragma unroll
    for (int ti = 0; ti < 9; ++ti) {
        int dy = ti / 3;
        int dx = ti % 3;
        const float* pa = &sH[dy][x0 + m + dx][0];
        #pragma unroll
        for (int kc = 0; kc < 4; ++kc) {         // 4-channel K chunks
            int k = 4 * kc + kb;
            v2f a;  a.x = pa[k];            a.y = pa[k + 1];
            v2f w;  w.x = sW2[ti][n][k];    w.y = sW2[ti][n][k + 1];
            acc = __builtin_amdgcn_wmma_f32_16x16x4_f32(
                false, a, false, w, (short)0, acc, false, false);
        }
    }

    if (n < COUT) {
        float bias = b2[n];
        int mtop = (lane >> 4) * 8;
        #pragma unroll
        for (int v = 0; v < 8; ++v) {
            int mm = mtop + v;
            gout[((size_t)(b * GH + y) * GW + (x0 + mm)) * COUT + n] = acc[v] + bias;
        }
    }
}

// ---------------------------------------------------------------------------
// 5) grid_to_points: bilinear gather, write (B, N, 4)
// ---------------------------------------------------------------------------
__global__ void fno_gather_kernel(const float* __restrict__ gout,
                                  const float* __restrict__ coords,
                                  float* __restrict__ out) {
    int idx = blockIdx.x * blockDim.x + threadIdx.x;   // 0 .. B*N-1
    if (idx >= BATCH * NPTS) return;
    int b = idx >> 18;
    float c0 = coords[2 * idx + 0];
    float c1 = coords[2 * idx + 1];
    // reference: x from coords[...,0], y from coords[...,1]; (2c-1+1)*0.5*(d-1) = c*(d-1)
    float x = c0 * (float)(GW - 1);
    float y = c1 * (float)(GH - 1);
    float x0f = floorf(x), y0f = floorf(y);
    float wx = x - x0f, wy = y - y0f;
    int x0 = min(max((int)x0f, 0), GW - 1);
    int x1 = min(x0 + 1, GW - 1);
    int y0 = min(max((int)y0f, 0), GH - 1);
    int y1 = min(y0 + 1, GH - 1);

    const float4* g = (const float4*)gout;
    size_t base = (size_t)b * GH * GW;
    float4 g00 = g[base + y0 * GW + x0];
    float4 g01 = g[base + y0 * GW + x1];
    float4 g10 = g[base + y1 * GW + x0];
    float4 g11 = g[base + y1 * GW + x1];

    float w00 = (1.0f - wx) * (1.0f - wy);
    float w01 = wx * (1.0f - wy);
    float w10 = (1.0f - wx) * wy;
    float w11 = wx * wy;

    float4 r;
    r.x = g00.x * w00 + g01.x * w01 + g10.x * w10 + g11.x * w11;
    r.y = g00.y * w00 + g01.y * w01 + g10.y * w10 + g11.y * w11;
    r.z = g00.z * w00 + g01.z * w01 + g10.z * w10 + g11.z * w11;
    r.w = g00.w * w00 + g01.w * w01 + g10.w * w10 + g11.w * w11;
    ((float4*)out)[idx] = r;
}

// ---------------------------------------------------------------------------
// Workspace layout (floats):
//   seg/grid_in : [0,            1048576)   B*16384*4
//   cnt         : [1048576,      1310720)   B*16384
//   hid         : [1310720,      5505024)   B*16384*16
//   gout        : [5505024,      6553600)   B*16384*4
// Total 26,214,400 bytes.
// ---------------------------------------------------------------------------
extern "C" void kernel_launch(void* const* d_in, const int* in_sizes, int n_in,
                              void* d_out, int out_size, void* d_ws, size_t ws_size,
                              hipStream_t stream) {
    const float* u      = (const float*)d_in[0];
    const float* coords = (const float*)d_in[1];
    const float* w1     = (const float*)d_in[2];
    const float* b1     = (const float*)d_in[3];
    const float* w2     = (const float*)d_in[4];
    const float* b2     = (const float*)d_in[5];
    float* out = (float*)d_out;
    float* ws  = (float*)d_ws;

    float* seg  = ws;
    float* cnt  = ws + 1048576;
    float* hid  = ws + 1310720;
    float* gout = ws + 5505024;

    const int nzero = 1310720;                        // seg + cnt contiguous
    fno_zero_kernel<<<(nzero + 255) / 256, 256, 0, stream>>>(seg, nzero);

    const int npts_total = BATCH * NPTS;              // 4,194,304
    fno_scatter_kernel<<<npts_total / 256, 256, 0, stream>>>(u, coords, seg, cnt);

    const int ncells = BATCH * GH * GW;               // 262,144
    fno_normalize_kernel<<<ncells / 256, 256, 0, stream>>>(seg, cnt);

    // one block per (batch,row): 16*128 = 2048 blocks, 8 waves (tiles) each
    fno_conv1_wmma_kernel<<<2048, 256, 0, stream>>>(seg, w1, b1, hid);
    fno_conv2_wmma_kernel<<<2048, 256, 0, stream>>>(hid, w2, b2, gout);

    fno_gather_kernel<<<npts_total / 256, 256, 0, stream>>>(gout, coords, out);
}